// SNNLayer_78743930404901
// MI455X (gfx1250) — compile-verified
//
#include <hip/hip_runtime.h>

typedef __attribute__((ext_vector_type(2))) float v2f;
typedef __attribute__((ext_vector_type(8))) float v8f;

constexpr int N     = 8192;  // graph nodes
constexpr int C     = 16;    // C_IN == C_OUT
constexpr int KTAPS = 5;     // polynomial taps
constexpr int WAVES = 8;     // waves per block (blockDim = 256)
constexpr int KCHUNK = N / WAVES;  // 1024 K per wave

// ---------------------------------------------------------------------------
// acc = relu(x @ W0)   (tiny: 8192x16x16)
// ---------------------------------------------------------------------------
__global__ void __launch_bounds__(256)
snn_init(const float* __restrict__ x, const float* __restrict__ w0,
         float* __restrict__ acc) {
  int idx = blockIdx.x * 256 + threadIdx.x;  // one output element each
  int r = idx >> 4, co = idx & 15;
  float s = 0.f;
#pragma unroll
  for (int ci = 0; ci < C; ++ci) s += x[r * C + ci] * w0[ci * C + co];
  acc[idx] = fmaxf(s, 0.f);
}

// ---------------------------------------------------------------------------
// One Laplacian tap:
//   xout = L @ xin      (one 16-row block per workgroup; K=8192 split 8 ways
//                        across waves, partial C tiles reduced through LDS)
//   acc += relu(xout @ Wp)   (+ optional final relu, fused into last tap)
// blockDim = 256 (8 waves); grid = N/16 = 512 blocks -> 4096 waves in flight.
// ---------------------------------------------------------------------------
__global__ void __launch_bounds__(256)
snn_pass(const float* __restrict__ L, const float* __restrict__ xin,
         const float* __restrict__ wp, float* __restrict__ xout,
         float* __restrict__ acc, int final_relu) {
  __shared__ float part[WAVES][16][17];  // per-wave partial tile (+1 pad)

  const int lane = threadIdx.x & 31;
  const int wave = threadIdx.x >> 5;
  const int row0 = blockIdx.x * 16;  // this block's 16-row tile of L
  const int m    = lane & 15;        // M index (A) / N index (B)
  const int h    = lane >> 4;        // lane half selects K pair
  const int k0   = wave * KCHUNK;    // this wave's K range

  // A frag (ISA 32-bit 16x4 layout): lane l -> L[row0+m][k + 2h + {0,1}]
  const float* Arow = L + (size_t)(row0 + m) * N + k0 + 2 * h;
  // B frag: lane l -> xin[k + 2h + {0,1}][m]
  const float* Bcol = xin + (size_t)(k0 + 2 * h) * C + m;

  v8f c = {};
  for (int k = 0; k < KCHUNK; k += 4) {
    if ((k & 63) == 0)  // stride-ahead prefetch of this lane's L row stream
      __builtin_prefetch(Arow + k + 256, 0, 0);
    v2f a = *(const v2f*)(Arow + k);  // 8B aligned: k%4==0, 2h even
    v2f b;
    b.x = Bcol[(size_t)k * C];
    b.y = Bcol[(size_t)k * C + C];
    c = __builtin_amdgcn_wmma_f32_16x16x4_f32(
        /*neg_a=*/false, a, /*neg_b=*/false, b,
        /*c_mod=*/(short)0, c, /*reuse_a=*/false, /*reuse_b=*/false);
  }

  // C/D layout: vgpr j, lanes 0-15 -> M=j, lanes 16-31 -> M=j+8; N = m.
#pragma unroll
  for (int j = 0; j < 8; ++j) part[wave][j + 8 * h][m] = c[j];
  __syncthreads();

  // Reduce the 8 partial tiles; thread t owns element (r, n) of the tile.
  const int r = threadIdx.x >> 4;
  const int n = threadIdx.x & 15;
  float v = 0.f;
#pragma unroll
  for (int w = 0; w < WAVES; ++w) v += part[w][r][n];
  xout[(size_t)(row0 + r) * C + n] = v;  // xp for the next tap (coalesced)
  part[0][r][n] = v;                     // same (r,n) this thread just read
  __syncthreads();

  // acc[row0+r][n] += relu( tile[r][:] @ Wp[:, n] )
  float s = 0.f;
#pragma unroll
  for (int ci = 0; ci < C; ++ci) s = fmaf(part[0][r][ci], wp[ci * C + n], s);
  size_t o = (size_t)(row0 + r) * C + n;
  float av = acc[o] + fmaxf(s, 0.f);
  acc[o] = final_relu ? fmaxf(av, 0.f) : av;
}

// ---------------------------------------------------------------------------
extern "C" void kernel_launch(void* const* d_in, const int* in_sizes, int n_in,
                              void* d_out, int out_size, void* d_ws,
                              size_t ws_size, hipStream_t stream) {
  const float* x = (const float*)d_in[0];        // [N, C]
  const float* L = (const float*)d_in[1];        // [N, N]
  const float* W = (const float*)d_in[2];        // [KTAPS, C, C]
  float* acc = (float*)d_out;                    // [N, C]
  float* xa  = (float*)d_ws;                     // ping
  float* xb  = xa + (size_t)N * C;               // pong

  snn_init<<<(N * C) / 256, 256, 0, stream>>>(x, W, acc);

  const float* cur = x;
  float* nxt = xa;
  for (int p = 1; p < KTAPS; ++p) {
    snn_pass<<<N / 16, 256, 0, stream>>>(
        L, cur, W + (size_t)p * C * C, nxt, acc, p == KTAPS - 1);
    cur = nxt;
    nxt = (nxt == xa) ? xb : xa;
  }
}